// CBOW_13709535608899
// MI455X (gfx1250) — compile-verified
//
#include <hip/hip_runtime.h>

// CBOW negative-sampling loss, MI455X (gfx1250, wave32).
//
// Phase 1 (bandwidth-bound, ~973 MB @ 23.3 TB/s ≈ 42 us floor):
//   one-hot rows are scanned with non-temporal GLOBAL_LOAD_B128 to find the
//   hot index (value 1.0f). One workgroup per row, 4864 rows.
// Phase 2 (tiny): gathered-embedding dot products done on the matrix pipe
//   with V_WMMA_F32_16X16X4_F32 (diagonal-of-outer-product trick), fp32 exact.
// Phase 3: deterministic tree reduction of per-(partner,batch) contributions.

#define VOC   50000
#define DIM   128
#define BATCH 256
#define CTX   8
#define KNEG  10
#define NROWS (BATCH + BATCH*CTX + BATCH*KNEG)   // 4864
#define NPART (KNEG + 1)                          // 1 vi_embed + 10 negatives

typedef float v2f __attribute__((ext_vector_type(2)));
typedef float v4f __attribute__((ext_vector_type(4)));
typedef float v8f __attribute__((ext_vector_type(8)));

// ---------------------------------------------------------------------------
// Phase 1: find the one-hot index of each row. Streaming, read-once data ->
// non-temporal 128-bit loads so we do not thrash L2 (192 MB << 973 MB).
// VOC = 50000 = 12500 float4s exactly, 16-byte aligned per row.
// ---------------------------------------------------------------------------
__global__ __launch_bounds__(256) void onehot_index_kernel(
    const float* __restrict__ vo, const float* __restrict__ vi,
    const float* __restrict__ neg, int* __restrict__ out_idx)
{
  const int row = blockIdx.x;
  const float* base;
  if (row < BATCH)                  base = vo  + (size_t)row * VOC;
  else if (row < BATCH + BATCH*CTX) base = vi  + (size_t)(row - BATCH) * VOC;
  else                              base = neg + (size_t)(row - BATCH - BATCH*CTX) * VOC;

  int found = -1;
  for (int i = threadIdx.x * 4; i < VOC; i += 256 * 4) {
    v4f v = __builtin_nontemporal_load((const v4f*)(base + i));
    if (v.x != 0.0f) found = i;
    if (v.y != 0.0f) found = i + 1;
    if (v.z != 0.0f) found = i + 2;
    if (v.w != 0.0f) found = i + 3;
  }
  // Exactly one lane in the block finds the hot element (inputs are one-hot).
  if (found >= 0) out_idx[row] = found;
}

// ---------------------------------------------------------------------------
// Phase 2: per-(tile of 16 batches, partner) wave computes
//   C = vo_embed_tile (16xDIM) x partner_tile^T (DIMx16)
// with V_WMMA_F32_16X16X4_F32, K-chunks of 4; diagonal C[i][i] is the wanted
// dot product.  One wave per block -> EXEC all ones at every WMMA.
//
// Fragment layout (ISA 7.12.2, 32-bit, 16x16x4):
//   A 16x4 : lane L(0..15) row M=L holds K={d,d+1} in v[0],v[1];
//            lane L+16 same row holds K={d+2,d+3}.
//   B 4x16 : mirrored by column N=lane&15 (rows striped across lanes/VGPR).
//   C 16x16: VGPR v, lanes 0..15 -> (M=v,   N=lane),
//                    lanes 16..31-> (M=v+8, N=lane-16).
//   Diagonal i<8  -> lane i,    c[i];  i>=8 -> lane i+16, c[i-8].
// ---------------------------------------------------------------------------
__device__ __forceinline__ float log_sigmoid(float x) {
  // stable: log_sigmoid(x) = min(x,0) - log1p(exp(-|x|))
  return fminf(x, 0.0f) - log1pf(expf(-fabsf(x)));
}

__global__ __launch_bounds__(32) void cbow_dot_kernel(
    const float* __restrict__ Vemb, const float* __restrict__ Uemb,
    const int* __restrict__ idx, float* __restrict__ contrib)
{
  const int lane = threadIdx.x;        // 0..31, full wave
  const int tile = blockIdx.x;         // 0..15  (16 batches each)
  const int p    = blockIdx.y;         // 0 = vi_embed, 1..10 = negatives

  const int r    = tile * 16 + (lane & 15);   // batch serviced by this lane's frag slice
  const int koff = (lane >> 4) * 2;           // K sub-offset within 4-chunk

  const int* idx_vo  = idx;
  const int* idx_vi  = idx + BATCH;
  const int* idx_neg = idx + BATCH + BATCH*CTX;

  const float* arow = Vemb + (size_t)idx_vo[r] * DIM;   // vo_embed row (gather)

  v8f acc = {};
  if (p == 0) {
    // partner = vi_embed = mean of 8 gathered U rows, folded into the B frag.
    const float* c0 = Uemb + (size_t)idx_vi[r*CTX + 0] * DIM;
    const float* c1 = Uemb + (size_t)idx_vi[r*CTX + 1] * DIM;
    const float* c2 = Uemb + (size_t)idx_vi[r*CTX + 2] * DIM;
    const float* c3 = Uemb + (size_t)idx_vi[r*CTX + 3] * DIM;
    const float* c4 = Uemb + (size_t)idx_vi[r*CTX + 4] * DIM;
    const float* c5 = Uemb + (size_t)idx_vi[r*CTX + 5] * DIM;
    const float* c6 = Uemb + (size_t)idx_vi[r*CTX + 6] * DIM;
    const float* c7 = Uemb + (size_t)idx_vi[r*CTX + 7] * DIM;
    for (int d = 0; d < DIM; d += 4) {
      v2f a = *(const v2f*)(arow + d + koff);
      v2f b = *(const v2f*)(c0 + d + koff) + *(const v2f*)(c1 + d + koff)
            + *(const v2f*)(c2 + d + koff) + *(const v2f*)(c3 + d + koff)
            + *(const v2f*)(c4 + d + koff) + *(const v2f*)(c5 + d + koff)
            + *(const v2f*)(c6 + d + koff) + *(const v2f*)(c7 + d + koff);
      b *= (1.0f / CTX);
      acc = __builtin_amdgcn_wmma_f32_16x16x4_f32(
          false, a, false, b, (short)0, acc, false, false);
    }
  } else {
    const float* brow = Uemb + (size_t)idx_neg[r*KNEG + (p - 1)] * DIM;
    for (int d = 0; d < DIM; d += 4) {
      v2f a = *(const v2f*)(arow + d + koff);
      v2f b = *(const v2f*)(brow + d + koff);
      acc = __builtin_amdgcn_wmma_f32_16x16x4_f32(
          false, a, false, b, (short)0, acc, false, false);
    }
  }

  // Diagonal extraction: lanes 0..7 own batches tile*16+0..7 (c[lane]);
  // lanes 16..23 own batches tile*16+8..15 (c[lane-16]).
  const float dotv = acc[lane & 7];
  if ((lane & 8) == 0) {
    const int b_owned = tile * 16 + (lane & 7) + ((lane >> 4) << 3);
    const float s = (p == 0) ? log_sigmoid(dotv) : log_sigmoid(-dotv);
    contrib[p * BATCH + b_owned] = s;   // deterministic: unique slot per (p, b)
  }
}

// ---------------------------------------------------------------------------
// Phase 3: deterministic reduction  out = mean_b( -(sum_p contrib[p][b]) )
// ---------------------------------------------------------------------------
__global__ __launch_bounds__(256) void cbow_reduce_kernel(
    const float* __restrict__ contrib, float* __restrict__ out)
{
  const int b = threadIdx.x;
  float s = 0.0f;
  for (int p = 0; p < NPART; ++p) s += contrib[p * BATCH + b];

  __shared__ float red[BATCH];
  red[b] = -s;
  __syncthreads();
  for (int off = BATCH / 2; off > 0; off >>= 1) {
    if (b < off) red[b] += red[b + off];
    __syncthreads();
  }
  if (b == 0) out[0] = red[0] * (1.0f / BATCH);
}

// ---------------------------------------------------------------------------
extern "C" void kernel_launch(void* const* d_in, const int* in_sizes, int n_in,
                              void* d_out, int out_size, void* d_ws, size_t ws_size,
                              hipStream_t stream) {
  const float* vo   = (const float*)d_in[0];   // [B, VOC]
  const float* vi   = (const float*)d_in[1];   // [B, CTX, VOC]
  const float* neg  = (const float*)d_in[2];   // [B, K, VOC]
  const float* Vemb = (const float*)d_in[3];   // [VOC, DIM]
  const float* Uemb = (const float*)d_in[4];   // [VOC, DIM]
  float* out = (float*)d_out;                  // scalar

  // Workspace: [NROWS] int indices, then [NPART*BATCH] float contributions.
  int*   idx     = (int*)d_ws;
  float* contrib = (float*)((char*)d_ws + NROWS * sizeof(int));

  onehot_index_kernel<<<dim3(NROWS), dim3(256), 0, stream>>>(vo, vi, neg, idx);
  cbow_dot_kernel<<<dim3(16, NPART), dim3(32), 0, stream>>>(Vemb, Uemb, idx, contrib);
  cbow_reduce_kernel<<<dim3(1), dim3(256), 0, stream>>>(contrib, out);
}